// GCN_17514876633977
// MI455X (gfx1250) — compile-verified
//
#include <hip/hip_runtime.h>

// ---------------------------------------------------------------------------
// GCN: h = ((.5A)^2 x) -> relu(.@W1+b1) -> ((.5A)^2 .) -> .@W2+b2
// SpMM: wave-per-edge gather + global_atomic_add_f32 scatter (L2-resident).
// Dense linears: V_WMMA_F32_16X16X4_F32 fed from LDS-staged tiles
// (async global->LDS copies when the toolchain exposes the builtin).
// ---------------------------------------------------------------------------

typedef __attribute__((ext_vector_type(2))) float v2f;
typedef __attribute__((ext_vector_type(8))) float v8f;

#if __has_builtin(__builtin_amdgcn_global_load_async_to_lds_b128)
#define GCN_ASYNC_LDS 1
typedef int gcn_v4i __attribute__((vector_size(16)));  // matches builtin param type
#endif

// ---- zero fill ------------------------------------------------------------
__global__ void gcn_zero(float* __restrict__ p, int n) {
    int i = blockIdx.x * blockDim.x + threadIdx.x;
    if (i < n) p[i] = 0.0f;
}

// ---- SpMM: out[r] += (0.5*val_e) * in[c]  (one wave per edge) -------------
// D = 32: 1 float per lane; D = 64: float2 per lane.
template <int D>
__global__ void gcn_spmm(const float* __restrict__ h_in,
                         const int* __restrict__ erow,
                         const int* __restrict__ ecol,
                         const float* __restrict__ eval,
                         float* __restrict__ h_out,
                         int n_edges) {
    int gid  = blockIdx.x * blockDim.x + threadIdx.x;
    int lane = gid & 31;
    int e    = gid >> 5;
    if (e >= n_edges) return;

    int   r = erow[e];
    int   c = ecol[e];
    float v = 0.5f * eval[e];

    if (D == 32) {
        float x = h_in[c * 32 + lane];
        atomicAdd(&h_out[r * 32 + lane], v * x);
    } else {
        const float2* in2 = reinterpret_cast<const float2*>(h_in);
        float2 x = in2[c * 32 + lane];
        atomicAdd(&h_out[r * 64 + 2 * lane + 0], v * x.x);
        atomicAdd(&h_out[r * 64 + 2 * lane + 1], v * x.y);
    }
}

// ---- Dense linear via WMMA f32 16x16x4, LDS-staged operands ---------------
// out[N,64] = act(in[N,K] @ W[K,64] + bias), K = 32 or 64 (row-major W).
// Block = 128 threads (4 waves); block -> 16-row tile, wave w -> cols 16w..16w+15.
// A 16x4 layout:  lane = 16*kh + m holds A[m][ka], A[m][ka+1], ka = kb+2*kh
// B 4x16 layout:  lane = 16*kh + n holds B[ka][n], B[ka+1][n]
// C/D 16x16:      vgpr r, lane = 16*kh + n holds C[r + 8*kh][n]
template <int K, bool RELU>
__global__ void gcn_linear_wmma(const float* __restrict__ in,
                                const float* __restrict__ W,
                                const float* __restrict__ bias,
                                float* __restrict__ out) {
    __shared__ float sA[16 * K];   // A tile: rows m0..m0+15 (contiguous in global)
    __shared__ float sW[K * 64];   // full weight matrix
    __shared__ float sB[64];       // bias

    const int tid = threadIdx.x;
    const int m0  = blockIdx.x * 16;

    // ---- cooperative staging: global -> LDS (float4 granularity) ----
    const float4* gA = (const float4*)(in + m0 * K);   // 4*K float4s
    const float4* gW = (const float4*)W;               // 16*K float4s
    const float4* gB = (const float4*)bias;            // 16 float4s
    float4* lA = (float4*)sA;
    float4* lW = (float4*)sW;
    float4* lB = (float4*)sB;

#ifdef GCN_ASYNC_LDS
    for (int i = tid; i < 4 * K; i += 128)
        __builtin_amdgcn_global_load_async_to_lds_b128(
            (gcn_v4i*)&gA[i], (gcn_v4i*)&lA[i], 0, 0);
    for (int i = tid; i < 16 * K; i += 128)
        __builtin_amdgcn_global_load_async_to_lds_b128(
            (gcn_v4i*)&gW[i], (gcn_v4i*)&lW[i], 0, 0);
    if (tid < 16)
        __builtin_amdgcn_global_load_async_to_lds_b128(
            (gcn_v4i*)&gB[tid], (gcn_v4i*)&lB[tid], 0, 0);
#if __has_builtin(__builtin_amdgcn_s_wait_asynccnt)
    __builtin_amdgcn_s_wait_asynccnt(0);
#else
    asm volatile("s_wait_asynccnt 0x0" ::: "memory");
#endif
#else
    for (int i = tid; i < 4 * K; i += 128)  lA[i] = gA[i];
    for (int i = tid; i < 16 * K; i += 128) lW[i] = gW[i];
    if (tid < 16) lB[tid] = gB[tid];
#endif
    __syncthreads();

    // ---- WMMA compute from LDS ----
    const int lane = tid & 31;
    const int wid  = tid >> 5;
    const int n0   = wid * 16;     // output-column tile
    const int m    = lane & 15;    // row (A) / col (B,C)
    const int kh   = lane >> 4;    // k-half select

    v8f acc = {};
    #pragma unroll
    for (int kb = 0; kb < K; kb += 4) {
        const int ka = kb + 2 * kh;
        v2f a, b;
        a.x = sA[m * K + ka];      // 8B-aligned pair -> ds_load_b64
        a.y = sA[m * K + ka + 1];
        b.x = sW[(ka    ) * 64 + n0 + m];
        b.y = sW[(ka + 1) * 64 + n0 + m];
        acc = __builtin_amdgcn_wmma_f32_16x16x4_f32(
            /*neg_a=*/false, a, /*neg_b=*/false, b,
            /*c_mod=*/(short)0, acc, /*reuse_a=*/false, /*reuse_b=*/false);
    }

    const float bv = sB[n0 + m];
    #pragma unroll
    for (int r = 0; r < 8; ++r) {
        float val = acc[r] + bv;
        if (RELU) val = fmaxf(val, 0.0f);
        out[(m0 + r + 8 * kh) * 64 + n0 + m] = val;
    }
}

// ---------------------------------------------------------------------------
extern "C" void kernel_launch(void* const* d_in, const int* in_sizes, int n_in,
                              void* d_out, int out_size, void* d_ws, size_t ws_size,
                              hipStream_t stream) {
    const float* x    = (const float*)d_in[0];
    const float* eval = (const float*)d_in[1];
    const int*   erow = (const int*)d_in[2];
    const int*   ecol = (const int*)d_in[3];
    const float* W1   = (const float*)d_in[4];
    const float* b1   = (const float*)d_in[5];
    const float* W2   = (const float*)d_in[6];
    const float* b2   = (const float*)d_in[7];

    float* outp = (float*)d_out;        // [N,64] final; reused as temp
    float* t0   = (float*)d_ws;         // [N,64] temp

    const int n_nodes = in_sizes[0] / 32;
    const int n_edges = in_sizes[1];

    const int ZB = 256;
    const int n32 = n_nodes * 32;
    const int n64 = n_nodes * 64;

    dim3 spmm_grid((unsigned)(((size_t)n_edges * 32 + 255) / 256));
    dim3 lin_grid((unsigned)(n_nodes / 16));

    // layer 1: propagate x twice at D=32
    gcn_zero<<<(n32 + ZB - 1) / ZB, ZB, 0, stream>>>(t0, n32);
    gcn_spmm<32><<<spmm_grid, 256, 0, stream>>>(x, erow, ecol, eval, t0, n_edges);
    gcn_zero<<<(n32 + ZB - 1) / ZB, ZB, 0, stream>>>(outp, n32);
    gcn_spmm<32><<<spmm_grid, 256, 0, stream>>>(t0, erow, ecol, eval, outp, n_edges);

    // h = relu(h @ W1 + b1): [N,32] -> [N,64]
    gcn_linear_wmma<32, true><<<lin_grid, 128, 0, stream>>>(outp, W1, b1, t0);

    // layer 2: propagate twice at D=64
    gcn_zero<<<(n64 + ZB - 1) / ZB, ZB, 0, stream>>>(outp, n64);
    gcn_spmm<64><<<spmm_grid, 256, 0, stream>>>(t0, erow, ecol, eval, outp, n_edges);
    gcn_zero<<<(n64 + ZB - 1) / ZB, ZB, 0, stream>>>(t0, n64);
    gcn_spmm<64><<<spmm_grid, 256, 0, stream>>>(outp, erow, ecol, eval, t0, n_edges);

    // out = h @ W2 + b2: [N,64] -> [N,64]
    gcn_linear_wmma<64, false><<<lin_grid, 128, 0, stream>>>(t0, W2, b2, outp);
}